// stagenet_39092792328412
// MI455X (gfx1250) — compile-verified
//
#include <hip/hip_runtime.h>
#include <math.h>
#include <stdint.h>

// Problem dims (fixed by setup_inputs)
#define B_   2
#define C_   32
#define H_   128
#define W_   160
#define D_   32
#define N_   4
#define G_   8
#define HW_  (H_*W_)
#define WAVES 8           // waves per block; 1 wave (32 lanes) = 1 pixel, lane = depth

// ---------------------------------------------------------------------------
// Kernel 1: per-(src,b) homography matrices.
// proj layout: (B, N+1, 2, 4, 4). For each view v: new = E with rows[:3] = K3x3 @ E[:3,:4].
// Output per (s,b): rot(3x3) row-major + trans(3) of  src_new @ inv(ref_new).
// Both matrices have last row [0,0,0,1] -> affine inverse via 3x3 adjugate.
// ---------------------------------------------------------------------------
__global__ void k_setup_mats(const float* __restrict__ proj, float* __restrict__ mats) {
    int t = threadIdx.x;
    if (t >= N_ * B_) return;
    int s = t / B_;
    int b = t - s * B_;
    const float* refE = proj + (((size_t)b * (N_ + 1) + 0) * 2 + 0) * 16;
    const float* refK = refE + 16;
    const float* srcE = proj + (((size_t)b * (N_ + 1) + (s + 1)) * 2 + 0) * 16;
    const float* srcK = srcE + 16;

    float RM[3][4], SM[3][4];
    for (int i = 0; i < 3; ++i)
        for (int j = 0; j < 4; ++j) {
            float a = 0.f, c = 0.f;
            for (int k = 0; k < 3; ++k) {
                a = fmaf(refK[i * 4 + k], refE[k * 4 + j], a);
                c = fmaf(srcK[i * 4 + k], srcE[k * 4 + j], c);
            }
            RM[i][j] = a; SM[i][j] = c;
        }

    float a00 = RM[0][0], a01 = RM[0][1], a02 = RM[0][2];
    float a10 = RM[1][0], a11 = RM[1][1], a12 = RM[1][2];
    float a20 = RM[2][0], a21 = RM[2][1], a22 = RM[2][2];
    float c00 =  (a11 * a22 - a12 * a21);
    float c01 = -(a10 * a22 - a12 * a20);
    float c02 =  (a10 * a21 - a11 * a20);
    float det = a00 * c00 + a01 * c01 + a02 * c02;
    float id  = 1.0f / det;
    float I[3][3];
    I[0][0] = c00 * id;  I[0][1] = -(a01 * a22 - a02 * a21) * id;  I[0][2] =  (a01 * a12 - a02 * a11) * id;
    I[1][0] = c01 * id;  I[1][1] =  (a00 * a22 - a02 * a20) * id;  I[1][2] = -(a00 * a12 - a02 * a10) * id;
    I[2][0] = c02 * id;  I[2][1] = -(a00 * a21 - a01 * a20) * id;  I[2][2] =  (a00 * a11 - a01 * a10) * id;

    float tb[3];
    for (int i = 0; i < 3; ++i)
        tb[i] = -(I[i][0] * RM[0][3] + I[i][1] * RM[1][3] + I[i][2] * RM[2][3]);

    float* o = mats + (size_t)(s * B_ + b) * 12;
    for (int i = 0; i < 3; ++i) {
        for (int j = 0; j < 3; ++j)
            o[i * 3 + j] = SM[i][0] * I[0][j] + SM[i][1] * I[1][j] + SM[i][2] * I[2][j];
        o[9 + i] = SM[i][0] * tb[0] + SM[i][1] * tb[1] + SM[i][2] * tb[2] + SM[i][3];
    }
}

// ---------------------------------------------------------------------------
// Kernel 2: transpose (M, C, HW) -> (M, HW, C) so bilinear taps are contiguous
// b128 loads. One thread per (m, hw): coalesced NT reads per channel (single
// use -> don't pollute L2's src-feature working set), RT wide stores (the
// gather kernel wants them L2-resident).
// ---------------------------------------------------------------------------
__global__ void k_transpose(const float* __restrict__ in, float* __restrict__ out, int M) {
    int t = blockIdx.x * blockDim.x + threadIdx.x;
    if (t >= M * HW_) return;
    int m  = t / HW_;
    int hw = t - m * HW_;
    const float* src = in + (size_t)m * C_ * HW_ + hw;
    float4* dst = (float4*)(out + (size_t)t * C_);
#pragma unroll
    for (int c0 = 0; c0 < C_; c0 += 4) {
        float4 v;
        v.x = __builtin_nontemporal_load(src + (size_t)(c0 + 0) * HW_);
        v.y = __builtin_nontemporal_load(src + (size_t)(c0 + 1) * HW_);
        v.z = __builtin_nontemporal_load(src + (size_t)(c0 + 2) * HW_);
        v.w = __builtin_nontemporal_load(src + (size_t)(c0 + 3) * HW_);
        dst[c0 >> 2] = v;
    }
}

// ---------------------------------------------------------------------------
// Kernel 3: main fused kernel. wave = pixel, lane = depth (wave32 == D).
// ---------------------------------------------------------------------------
__global__ void __launch_bounds__(WAVES * 32)
k_main(const float* __restrict__ refT,    // (B*HW, C)
       const float* __restrict__ srcT,    // (N*B*HW, C)
       const float* __restrict__ depthv,  // (B, D, HW)
       const float* __restrict__ mats,    // (N*B, 12)
       const float* __restrict__ regw,    // (8)
       const float* __restrict__ regb,    // (1)
       float* __restrict__ out_depth,     // (B*HW)
       float* __restrict__ out_attn)      // (B, D, HW)
{
    __shared__ __align__(16) float lds_ref[WAVES * C_];
    const int tid  = threadIdx.x;
    const int wave = tid >> 5;
    const int lane = tid & 31;                 // depth index d
    const int p0   = blockIdx.x * WAVES;       // first pixel of this block
    const int p    = p0 + wave;

    // ---- stage ref-feature tile (8 px * 32 ch = 1KB, contiguous) via CDNA5
    // async global->LDS copy (ASYNCcnt path, CDNA5 ISA). 64 lanes x b128.
    if (tid < 64) {
        uint64_t gaddr = (uint64_t)(uintptr_t)(refT + (size_t)p0 * C_ + (size_t)tid * 4);
        uint32_t laddr = (uint32_t)(uintptr_t)(&lds_ref[tid * 4]);
        asm volatile("global_load_async_to_lds_b128 %0, %1, off"
                     :: "v"(laddr), "v"(gaddr) : "memory");
    }
    asm volatile("s_wait_asynccnt 0" ::: "memory");
    __syncthreads();

    // b is block-uniform (HW_ % WAVES == 0) -> derive from blockIdx so that
    // matrix loads below become SMEM (s_load) instead of per-lane VMEM.
    const int b  = p0 / HW_;
    const int hw = p - b * HW_;
    const int h  = hw / W_;
    const int w  = hw - h * W_;

    const float dval = depthv[((size_t)b * D_ + lane) * HW_ + hw];

    float feats[G_];
#pragma unroll
    for (int g = 0; g < G_; ++g) feats[g] = 0.f;
    float cw = 1e-8f;
    const float inv_sqrtC = 0.17677669529663687f;   // 1/sqrt(32)

    const float4* rf = (const float4*)(lds_ref + wave * C_);

#pragma unroll
    for (int s = 0; s < N_; ++s) {
        const float* M = mats + (size_t)(s * B_ + b) * 12;   // scalar address -> s_load
        const float fx = (float)w, fy = (float)h;
        const float rx = M[0] * fx + M[1] * fy + M[2];
        const float ry = M[3] * fx + M[4] * fy + M[5];
        const float rz = M[6] * fx + M[7] * fy + M[8];

        float px = fmaf(rx, dval, M[9]);
        float py = fmaf(ry, dval, M[10]);
        float pz = fmaf(rz, dval, M[11]);
        pz = (pz == 0.f) ? 1e-9f : pz;
        const float gx = px / pz;
        const float gy = py / pz;

        const float x0f = floorf(gx), y0f = floorf(gy);
        const float x1f = x0f + 1.f,  y1f = y0f + 1.f;
        const float wx = gx - x0f, wy = gy - y0f;

        const float inx0 = (x0f >= 0.f && x0f <= (float)(W_ - 1)) ? 1.f : 0.f;
        const float inx1 = (x1f >= 0.f && x1f <= (float)(W_ - 1)) ? 1.f : 0.f;
        const float iny0 = (y0f >= 0.f && y0f <= (float)(H_ - 1)) ? 1.f : 0.f;
        const float iny1 = (y1f >= 0.f && y1f <= (float)(H_ - 1)) ? 1.f : 0.f;

        float w00 = (1.f - wx) * (1.f - wy) * (inx0 * iny0);
        float w10 = wx         * (1.f - wy) * (inx1 * iny0);
        float w01 = (1.f - wx) * wy         * (inx0 * iny1);
        float w11 = wx         * wy         * (inx1 * iny1);

        const int xi0 = (int)fminf(fmaxf(x0f, 0.f), (float)(W_ - 1));
        const int xi1 = (int)fminf(fmaxf(x1f, 0.f), (float)(W_ - 1));
        const int yi0 = (int)fminf(fmaxf(y0f, 0.f), (float)(H_ - 1));
        const int yi1 = (int)fminf(fmaxf(y1f, 0.f), (float)(H_ - 1));

        const float* sb = srcT + (size_t)(s * B_ + b) * HW_ * C_;
        const float4* t00 = (const float4*)(sb + (size_t)(yi0 * W_ + xi0) * C_);
        const float4* t10 = (const float4*)(sb + (size_t)(yi0 * W_ + xi1) * C_);
        const float4* t01 = (const float4*)(sb + (size_t)(yi1 * W_ + xi0) * C_);
        const float4* t11 = (const float4*)(sb + (size_t)(yi1 * W_ + xi1) * C_);
        __builtin_prefetch(t00, 0, 0);   // global_prefetch_b8
        __builtin_prefetch(t11, 0, 0);

        float cf[G_];
#pragma unroll
        for (int g = 0; g < G_; ++g) cf[g] = 0.f;

#pragma unroll
        for (int c4 = 0; c4 < C_ / 4; ++c4) {        // group g == c4 (C/G == 4)
            const float4 a = t00[c4], q = t10[c4], u = t01[c4], v = t11[c4];
            const float4 r = rf[c4];
            float sx = a.x * w00 + q.x * w10 + u.x * w01 + v.x * w11;
            float sy = a.y * w00 + q.y * w10 + u.y * w01 + v.y * w11;
            float sz = a.z * w00 + q.z * w10 + u.z * w01 + v.z * w11;
            float sw = a.w * w00 + q.w * w10 + u.w * w01 + v.w * w11;
            cf[c4] = fmaf(sx, r.x, fmaf(sy, r.y, fmaf(sz, r.z, fmaf(sw, r.w, cf[c4]))));
        }

        float ssum = 0.f;
#pragma unroll
        for (int g = 0; g < G_; ++g) { cf[g] *= 0.25f; ssum += cf[g]; }  // mean over C/G

        // softmax over depth (wave lanes), logit = sum_g cf / attn_temp(=2)
        float logit = ssum * 0.5f;
        float mx = logit;
#pragma unroll
        for (int off = 16; off >= 1; off >>= 1) mx = fmaxf(mx, __shfl_xor(mx, off, 32));
        float e = expf(logit - mx);
        float Z = e;
#pragma unroll
        for (int off = 16; off >= 1; off >>= 1) Z += __shfl_xor(Z, off, 32);
        const float wgt = (e / Z) * inv_sqrtC;

        cw += wgt;
#pragma unroll
        for (int g = 0; g < G_; ++g) feats[g] = fmaf(wgt, cf[g], feats[g]);
    }

    // normalize + regression head + final softmax over depth
    const float icw = 1.f / cw;
    float logit2 = regb[0];
#pragma unroll
    for (int g = 0; g < G_; ++g) logit2 = fmaf(feats[g] * icw, regw[g], logit2);

    float mx2 = logit2;
#pragma unroll
    for (int off = 16; off >= 1; off >>= 1) mx2 = fmaxf(mx2, __shfl_xor(mx2, off, 32));
    float e2 = expf(logit2 - mx2);
    float Z2 = e2;
#pragma unroll
    for (int off = 16; off >= 1; off >>= 1) Z2 += __shfl_xor(Z2, off, 32);
    const float attn = e2 / Z2;

    out_attn[((size_t)b * D_ + lane) * HW_ + hw] = attn;

    // argmax over depth, first-max tie-break (matches jnp.argmax)
    float bv = attn; int bi = lane;
#pragma unroll
    for (int off = 16; off >= 1; off >>= 1) {
        float ov = __shfl_xor(bv, off, 32);
        int   oi = __shfl_xor(bi, off, 32);
        if (ov > bv || (ov == bv && oi < bi)) { bv = ov; bi = oi; }
    }
    if (lane == bi) out_depth[p] = dval;
}

// ---------------------------------------------------------------------------
extern "C" void kernel_launch(void* const* d_in, const int* in_sizes, int n_in,
                              void* d_out, int out_size, void* d_ws, size_t ws_size,
                              hipStream_t stream) {
    const float* ref   = (const float*)d_in[0];   // (B,C,H,W)
    const float* src   = (const float*)d_in[1];   // (N,B,C,H,W)
    const float* proj  = (const float*)d_in[2];   // (B,N+1,2,4,4)
    const float* depth = (const float*)d_in[3];   // (B,D,H,W)
    const float* regw  = (const float*)d_in[4];   // (8)
    const float* regb  = (const float*)d_in[5];   // (1)

    float* out_depth = (float*)d_out;                       // (B*HW)
    float* out_attn  = out_depth + (size_t)B_ * HW_;        // (B,D,HW)

    float* srcT = (float*)d_ws;                                      // (N*B*HW, C)
    float* refT = srcT + (size_t)N_ * B_ * HW_ * C_;                 // (B*HW, C)
    float* mats = refT + (size_t)B_ * HW_ * C_;                      // (N*B, 12)

    k_setup_mats<<<1, 32, 0, stream>>>(proj, mats);

    {
        int n = N_ * B_ * HW_;
        k_transpose<<<(n + 255) / 256, 256, 0, stream>>>(src, srcT, N_ * B_);
    }
    {
        int n = B_ * HW_;
        k_transpose<<<(n + 255) / 256, 256, 0, stream>>>(ref, refT, B_);
    }

    int nblocks = (B_ * HW_) / WAVES;   // 40960 / 8 = 5120
    k_main<<<nblocks, WAVES * 32, 0, stream>>>(refT, srcT, depth, mats,
                                               regw, regb, out_depth, out_attn);
}